// DyanEncoder_29583734735161
// MI455X (gfx1250) — compile-verified
//
#include <hip/hip_runtime.h>
#include <math.h>

// ---------------- problem constants (shapes fixed by the reference) --------
#define TT      36          // time steps
#define NP      160         // poles
#define KK      641         // dictionary atoms = 1 + 4*NP
#define KP      656         // K padded to 41*16 (WMMA M/N tile = 16, K-step = 4)
#define PP      512         // pixels
#define MT      41          // KP/16 M-tiles
#define NITER   200         // FISTA iterations
#define LAM     0.1f
#define EPSW    0.01f

#define NWAVE   16          // waves per fused-FISTA block (512 threads)
#define TPW     3           // max M-tiles per wave: ceil(41/16)

typedef float v2f __attribute__((ext_vector_type(2)));
typedef float v8f __attribute__((ext_vector_type(8)));

// ---------------- setup kernels -------------------------------------------
__global__ void dyan_build_dict(const float* __restrict__ rr,
                                const float* __restrict__ th,
                                float* __restrict__ D,        // [TT][KP]
                                float* __restrict__ dic_out)  // [TT][KK]
{
    int idx = blockIdx.x * blockDim.x + threadIdx.x;
    if (idx >= TT * KP) return;
    int t = idx / KP, k = idx % KP;
    float val = 0.f;
    if (k == 0) {
        val = 1.f;
    } else if (k < KK) {
        int g = (k - 1) / NP, j = (k - 1) % NP;
        float r = rr[j], a = th[j];
        float pr  = powf(r, (float)t);
        float sgn = (t & 1) ? -1.f : 1.f;
        float c = cosf((float)t * a), s = sinf((float)t * a);
        val = (g == 0) ? pr * c : (g == 1) ? sgn * pr * c
            : (g == 2) ? pr * s : sgn * pr * s;
    }
    D[idx] = val;
    if (k < KK) dic_out[t * KK + k] = val;
}

__global__ void dyan_dtd(const float* __restrict__ D, float* __restrict__ dtd)
{
    int idx = blockIdx.x * blockDim.x + threadIdx.x;
    if (idx >= KP * KP) return;
    int i = idx / KP, j = idx % KP;
    float s = 0.f;
    if (i < KK && j < KK)
        for (int t = 0; t < TT; ++t) s += D[t * KP + i] * D[t * KP + j];
    dtd[idx] = s;
}

// power iteration for the spectral norm of symmetric PSD DtD -> scal[0]=1/L
__global__ __launch_bounds__(256) void dyan_power(const float* __restrict__ dtd,
                                                  float* __restrict__ scal)
{
    __shared__ float v[KP], u[KP], red[256];
    int tid = threadIdx.x;
    for (int r = tid; r < KP; r += 256) v[r] = (r < KK) ? 1.f : 0.f;
    __syncthreads();
    float nrm = 1.f;
    for (int iter = 0; iter < 64; ++iter) {
        for (int r = tid; r < KP; r += 256) {
            const float* row = dtd + r * KP;
            float s = 0.f;
            for (int j = 0; j < KP; ++j) s += row[j] * v[j];
            u[r] = s;
        }
        __syncthreads();
        float ls = 0.f;
        for (int r = tid; r < KP; r += 256) ls += u[r] * u[r];
        red[tid] = ls; __syncthreads();
        for (int off = 128; off > 0; off >>= 1) {
            if (tid < off) red[tid] += red[tid + off];
            __syncthreads();
        }
        nrm = sqrtf(red[0]);
        float inv = 1.f / nrm;
        for (int r = tid; r < KP; r += 256) v[r] = u[r] * inv;
        __syncthreads();
    }
    if (tid == 0) scal[0] = 1.f / nrm;   // Linv
}

__global__ void dyan_a(const float* __restrict__ dtd,
                       const float* __restrict__ scal,
                       float* __restrict__ A)
{
    int idx = blockIdx.x * blockDim.x + threadIdx.x;
    if (idx >= KP * KP) return;
    int i = idx / KP, j = idx % KP;
    float Linv = scal[0];
    float a = 0.f;
    if (i < KK && j < KK) a = ((i == j) ? 1.f : 0.f) - dtd[idx] * Linv;
    A[idx] = a;
}

__global__ void dyan_dty(const float* __restrict__ D,
                         const float* __restrict__ Y,   // [TT][PP]
                         float* __restrict__ dty)       // [KP][PP]
{
    int idx = blockIdx.x * blockDim.x + threadIdx.x;
    if (idx >= KP * PP) return;
    int k = idx / PP, p = idx % PP;
    float s = 0.f;
    if (k < KK)
        for (int t = 0; t < TT; ++t) s += D[t * KP + k] * Y[t * PP + p];
    dty[idx] = s;
}

// ---------------- reweight kernels ----------------------------------------
__global__ void dyan_c_round0(const float* __restrict__ dty,
                              const float* __restrict__ scal,
                              float* __restrict__ cm, float* __restrict__ cp)
{
    int idx = blockIdx.x * blockDim.x + threadIdx.x;
    if (idx >= KP * PP) return;
    int k = idx / PP;
    float Linv = scal[0];
    float d  = (k < KK) ? dty[idx] * Linv : 0.f;
    float wl = (k < KK) ? LAM * Linv : 0.f;     // w == 1 in round 0
    cm[idx] = d - wl;
    cp[idx] = d + wl;
}

__global__ void dyan_zero_scalar(float* p) { *p = 0.f; }

__global__ __launch_bounds__(256) void dyan_wr_sumsq(const float* __restrict__ x,
                                                     float* __restrict__ sumsq)
{
    __shared__ float red[256];
    int tid = threadIdx.x;
    int idx = blockIdx.x * blockDim.x + tid;
    float val = 0.f;
    if (idx < KK * PP) {                        // rows 0..640 are contiguous
        float wr = 1.f / (fabsf(x[idx]) + EPSW);
        val = wr * wr;
    }
    red[tid] = val; __syncthreads();
    for (int off = 128; off > 0; off >>= 1) {
        if (tid < off) red[tid] += red[tid + off];
        __syncthreads();
    }
    if (tid == 0) atomicAdd(sumsq, red[0]);
}

__global__ void dyan_c_reweight(const float* __restrict__ dty,
                                const float* __restrict__ x,
                                const float* __restrict__ scal,
                                float* __restrict__ cm, float* __restrict__ cp)
{
    int idx = blockIdx.x * blockDim.x + threadIdx.x;
    if (idx >= KP * PP) return;
    int k = idx / PP;
    float Linv = scal[0];
    float inv_norm = rsqrtf(scal[1]);
    float d = 0.f, wl = 0.f;
    if (k < KK) {
        float wr = 1.f / (fabsf(x[idx]) + EPSW);
        float w  = wr * inv_norm;
        d  = dty[idx] * Linv;
        wl = w * LAM * Linv;
    }
    cm[idx] = d - wl;
    cp[idx] = d + wl;
}

// ---------------- fused FISTA round (WMMA f32 16x16x4) --------------------
// One block owns a 16-column slab of y/x for the whole 200-iteration round.
// y ping-pongs in LDS (2 x 656x16 f32 = 84 KB of the 320 KB WGP LDS);
// x and c+/- live entirely in VGPRs (compile-time-indexed arrays, fixed
// unrolled tile loops -> no scratch spills); only A streams from L2.
__global__ __launch_bounds__(NWAVE * 32) void dyan_fista_round(
    const float* __restrict__ A,     // [KP][KP], zero-padded
    const float* __restrict__ cm_g,  // [KP][PP]
    const float* __restrict__ cp_g,  // [KP][PP]
    float* __restrict__ x_g)         // [KP][PP]
{
    __shared__ float ybuf[2][KP * 16];

    const int tid   = threadIdx.x;
    const int lane  = tid & 31;
    const int wave  = tid >> 5;          // 16 waves
    const int n0    = blockIdx.x * 16;   // 32 blocks cover PP=512
    const int col   = lane & 15;         // N index / A-frag M index
    const int khalf = (lane >= 16) ? 2 : 0;
    const int mbase = (lane >= 16) ? 8 : 0;

    // zero initial y
    for (int i = tid; i < KP * 16; i += NWAVE * 32) ybuf[0][i] = 0.f;

    // preload c+/- and init x in registers (constant-indexed, no spills)
    float xreg[TPW][8], cmr[TPW][8], cpr[TPW][8];
    #pragma unroll
    for (int i = 0; i < TPW; ++i) {
        const int t = wave + NWAVE * i;          // wave-uniform
        if (t < MT) {
            const int m0 = t * 16;
            #pragma unroll
            for (int v = 0; v < 8; ++v) {
                int gi = (m0 + mbase + v) * PP + n0 + col;
                cmr[i][v] = cm_g[gi];
                cpr[i][v] = cp_g[gi];
                xreg[i][v] = 0.f;
            }
        } else {
            #pragma unroll
            for (int v = 0; v < 8; ++v) {
                cmr[i][v] = 0.f; cpr[i][v] = 0.f; xreg[i][v] = 0.f;
            }
        }
    }
    __syncthreads();

    float t_k = 1.f;
    for (int it = 0; it < NITER; ++it) {
        const float* ycur  = ybuf[it & 1];
        float*       ynext = ybuf[(it + 1) & 1];
        float t_new = 0.5f * (1.f + sqrtf(1.f + 4.f * t_k * t_k));
        float beta  = (t_k - 1.f) / t_new;
        t_k = t_new;

        #pragma unroll
        for (int i = 0; i < TPW; ++i) {
            const int t = wave + NWAVE * i;      // wave-uniform guard:
            if (t < MT) {                        // EXEC stays all-1s for WMMA
                const int m0 = t * 16;
                v8f acc = {};
                const float* arow = A + (m0 + col) * KP;   // A-frag: M = lane&15
                for (int kk = 0; kk < KP; kk += 4) {
                    int kb = kk + khalf;                   // lanes 16-31: K2/K3
                    v2f a; a.x = arow[kb];            a.y = arow[kb + 1];
                    v2f b; b.x = ycur[kb * 16 + col]; b.y = ycur[(kb + 1) * 16 + col];
                    acc = __builtin_amdgcn_wmma_f32_16x16x4_f32(
                            false, a, false, b, (short)0, acc, false, false);
                }
                #pragma unroll
                for (int v = 0; v < 8; ++v) {
                    float ay = acc[v];
                    float xn = fmaxf(0.f, ay + cmr[i][v]) + fminf(0.f, ay + cpr[i][v]);
                    float yn = xn + beta * (xn - xreg[i][v]);
                    xreg[i][v] = xn;
                    ynext[(m0 + mbase + v) * 16 + col] = yn;
                }
            }
        }
        __syncthreads();
    }

    // write final x for this round
    #pragma unroll
    for (int i = 0; i < TPW; ++i) {
        const int t = wave + NWAVE * i;
        if (t < MT) {
            const int m0 = t * 16;
            #pragma unroll
            for (int v = 0; v < 8; ++v)
                x_g[(m0 + mbase + v) * PP + n0 + col] = xreg[i][v];
        }
    }
}

__global__ void dyan_copy_final(const float* __restrict__ x, float* __restrict__ out)
{
    int idx = blockIdx.x * blockDim.x + threadIdx.x;
    if (idx < KK * PP) out[idx] = x[idx];
}

// ---------------- host-side launch ----------------------------------------
extern "C" void kernel_launch(void* const* d_in, const int* in_sizes, int n_in,
                              void* d_out, int out_size, void* d_ws, size_t ws_size,
                              hipStream_t stream)
{
    const float* X  = (const float*)d_in[0];  // [1,36,512]
    const float* rr = (const float*)d_in[1];  // [160]
    const float* th = (const float*)d_in[2];  // [160]
    float* out = (float*)d_out;               // final[641*512] ++ dic[36*641]
    float* ws  = (float*)d_ws;

    float* D    = ws;                       // TT*KP
    float* DtD  = D    + TT * KP;           // KP*KP
    float* Amat = DtD  + KP * KP;           // KP*KP
    float* DtY  = Amat + KP * KP;           // KP*PP
    float* cm   = DtY  + KP * PP;           // KP*PP
    float* cp   = cm   + KP * PP;           // KP*PP
    float* xg   = cp   + KP * PP;           // KP*PP
    float* scal = xg   + KP * PP;           // [0]=Linv, [1]=sum(wr^2)

    const int TB = 256;
    dyan_build_dict<<<(TT * KP + TB - 1) / TB, TB, 0, stream>>>(rr, th, D, out + KK * PP);
    dyan_dtd      <<<(KP * KP + TB - 1) / TB, TB, 0, stream>>>(D, DtD);
    dyan_power    <<<1, TB, 0, stream>>>(DtD, scal);
    dyan_a        <<<(KP * KP + TB - 1) / TB, TB, 0, stream>>>(DtD, scal, Amat);
    dyan_dty      <<<(KP * PP + TB - 1) / TB, TB, 0, stream>>>(D, X, DtY);

    for (int r = 0; r < 3; ++r) {
        if (r == 0) {
            dyan_c_round0<<<(KP * PP + TB - 1) / TB, TB, 0, stream>>>(DtY, scal, cm, cp);
        } else {
            dyan_zero_scalar<<<1, 1, 0, stream>>>(scal + 1);
            dyan_wr_sumsq<<<(KK * PP + TB - 1) / TB, TB, 0, stream>>>(xg, scal + 1);
            dyan_c_reweight<<<(KP * PP + TB - 1) / TB, TB, 0, stream>>>(DtY, xg, scal, cm, cp);
        }
        dyan_fista_round<<<PP / 16, NWAVE * 32, 0, stream>>>(Amat, cm, cp, xg);
    }

    dyan_copy_final<<<(KK * PP + TB - 1) / TB, TB, 0, stream>>>(xg, out);
}